// Node_8289286881404
// MI455X (gfx1250) — compile-verified
//
#include <hip/hip_runtime.h>
#include <stdint.h>

// Problem constants (from the reference): [B, X, Y, Z] fp32 grids.
#define BB 8
#define NX 128
#define NY 128
#define NZ 128

// Tile: TX x TY pencils of full Z per 256-thread block.
#define TX 8
#define TY 8
#define HX (TX + 2)
#define HY (TY + 2)

// Exact pointee type the builtin expects: vector of 4 ints (16 bytes).
typedef int v4i __attribute__((vector_size(16)));
typedef __attribute__((address_space(1))) v4i glob_v4i;  // global / "__device__"
typedef __attribute__((address_space(3))) v4i lds_v4i;   // LDS / "__shared__"

// Async global -> LDS copy of 16 bytes (GLOBAL_LOAD_ASYNC_TO_LDS_B128,
// tracked by ASYNCcnt). Builtin if the toolchain has it, inline asm otherwise.
__device__ __forceinline__ void async_g2l_b128(const float* gsrc, unsigned lds_byte_off) {
#if __has_builtin(__builtin_amdgcn_global_load_async_to_lds_b128)
  __builtin_amdgcn_global_load_async_to_lds_b128(
      (glob_v4i*)(uintptr_t)gsrc, (lds_v4i*)(uintptr_t)lds_byte_off,
      /*imm offset=*/0, /*cpol=*/0);
#else
  asm volatile("global_load_async_to_lds_b128 %0, %1, off"
               :
               : "v"(lds_byte_off), "v"(gsrc)
               : "memory");
#endif
}

__device__ __forceinline__ void wait_async0() {
#if __has_builtin(__builtin_amdgcn_s_wait_asynccnt)
  __builtin_amdgcn_s_wait_asynccnt(0);
#else
  asm volatile("s_wait_asynccnt 0x0" ::: "memory");
#endif
}

__global__ __launch_bounds__(256) void lap7_periodic_kernel(
    const float* __restrict__ mu, const float* __restrict__ act,
    const float* __restrict__ dxp, float* __restrict__ out) {
  // 10 x 10 x 128 fp32 halo tiles for mu and active: 2 * 51200 B = 100 KB LDS.
  __shared__ float smu[HX][HY][NZ];
  __shared__ float sac[HX][HY][NZ];

  const int tid = threadIdx.x;
  const int x0 = blockIdx.x * TX;
  const int y0 = blockIdx.y * TY;
  const int b = blockIdx.z;
  const size_t base_b = (size_t)b * NX * NY * NZ;

  const unsigned smu_base = (unsigned)(uintptr_t)&smu[0][0][0];
  const unsigned sac_base = (unsigned)(uintptr_t)&sac[0][0][0];

  // ---- Fill phase: HX*HY z-lines, 32 b128 segments per line, both arrays.
  const int nseg = NZ / 4;              // 32 float4 per line
  const int total = HX * HY * nseg;     // 3200
  for (int idx = tid; idx < total; idx += 256) {
    const int line = idx / nseg;
    const int seg = idx - line * nseg;
    const int li = line / HY;           // 0..HX-1
    const int lj = line - li * HY;      // 0..HY-1
    const int gx = (x0 + li - 1) & (NX - 1);   // periodic wrap
    const int gy = (y0 + lj - 1) & (NY - 1);
    const size_t goff = base_b + (((size_t)gx * NY + gy) * NZ) + (size_t)seg * 4;
    const unsigned loff = (unsigned)(line * NZ + seg * 4) * 4u;  // bytes
    async_g2l_b128(mu + goff, smu_base + loff);
    async_g2l_b128(act + goff, sac_base + loff);
  }
  wait_async0();
  __syncthreads();

  const float dxv = dxp[0];
  const float inv = 1.0f / (dxv * dxv);

  // ---- Compute phase: TX*TY lines x NZ points; lanes walk consecutive z.
  const int nout = TX * TY * NZ;        // 8192
  for (int k = tid; k < nout; k += 256) {
    const int oz = k & (NZ - 1);
    const int line = k >> 7;            // k / NZ
    const int ox = line / TY;
    const int oy = line - ox * TY;
    const int i = ox + 1;
    const int j = oy + 1;
    const int zm = (oz + NZ - 1) & (NZ - 1);
    const int zp = (oz + 1) & (NZ - 1);

    const float a_c = sac[i][j][oz];
    const float ec = smu[i][j][oz] * a_c;
    float s = smu[i - 1][j][oz] * sac[i - 1][j][oz];
    s += smu[i + 1][j][oz] * sac[i + 1][j][oz];
    s += smu[i][j - 1][oz] * sac[i][j - 1][oz];
    s += smu[i][j + 1][oz] * sac[i][j + 1][oz];
    s += smu[i][j][zm] * sac[i][j][zm];
    s += smu[i][j][zp] * sac[i][j][zp];

    const float lap = (s - 6.0f * ec) * inv;
    out[base_b + (((size_t)(x0 + ox) * NY + (y0 + oy)) * NZ) + oz] = lap * a_c;
  }
}

extern "C" void kernel_launch(void* const* d_in, const int* in_sizes, int n_in,
                              void* d_out, int out_size, void* d_ws, size_t ws_size,
                              hipStream_t stream) {
  (void)in_sizes; (void)n_in; (void)out_size; (void)d_ws; (void)ws_size;
  const float* mu = (const float*)d_in[0];
  const float* act = (const float*)d_in[1];
  const float* dx = (const float*)d_in[2];
  float* out = (float*)d_out;

  dim3 grid(NX / TX, NY / TY, BB);  // 16 x 16 x 8 = 2048 blocks
  lap7_periodic_kernel<<<grid, 256, 0, stream>>>(mu, act, dx, out);
}